// MeanConv_23021024707113
// MI455X (gfx1250) — compile-verified
//
#include <hip/hip_runtime.h>

// ---------------------------------------------------------------------------
// Fused multi-scale (k=3..15) box-mean + mask for 4096x4096 fp32, MI455X.
//   - Vertical: per-thread rolling 16-reg window, 7 running sums (add+sub).
//   - Horizontal: weighted suffix sums T_m in LDS; compiler pairs the +/-j
//     reads into ds_load_2addr_b32.
//   - 2 rows per barrier (halves sync overhead vs 1 row/barrier).
//   - mask rows staged 2 ahead via global_load_async_to_lds_b32 (ASYNCcnt),
//     x stream prefetched via global_prefetch_b8.
// ---------------------------------------------------------------------------

#define WIDTH   4096
#define HEIGHT  4096
#define RH      128     // rows processed per block
#define NTHR    256     // threads per block (= columns incl. 7+7 halo)
#define OUTW    242     // output columns per block
#define TS      272     // padded T-plane stride (floats), conflict-free

__device__ __forceinline__ int clampi(int v, int lo, int hi) {
    v = v < lo ? lo : v;
    return v > hi ? hi : v;
}

// Async copy of one fp32 per lane: global (SADDR base + per-lane byte offset)
// -> LDS. Tracked with ASYNCcnt. VDST carries the 32-bit LDS byte address.
__device__ __forceinline__ void async_f32_to_lds(const float* base_row,
                                                 unsigned vbyteoff,
                                                 float*   lds_dst) {
    unsigned lds_addr = (unsigned)(unsigned long long)(size_t)lds_dst;
    asm volatile("global_load_async_to_lds_b32 %0, %1, %2"
                 :
                 : "v"(lds_addr), "v"(vbyteoff), "s"(base_row)
                 : "memory");
}

#define WAIT_ASYNCCNT(n) asm volatile("s_wait_asynccnt " #n ::: "memory")

__global__ void __launch_bounds__(NTHR)
msbox7_kernel(const float* __restrict__ x,
              const float* __restrict__ mask,
              float* __restrict__ out)
{
    // [step parity][row-in-step][plane][column]
    __shared__ float Tlds[2][2][7][TS];
    __shared__ float Mlds[2][2][NTHR];

    const int tid = (int)threadIdx.x;
    const int c0  = OUTW * (int)blockIdx.x;
    const int cg  = c0 + tid - 7;                 // global column (may be OOB)
    const int cgc = clampi(cg, 0, WIDTH - 1);     // clamped = replicate pad
    const int r0  = RH * (int)blockIdx.y;         // multiple of 16
    const int rmax = r0 + RH - 1;
    const unsigned mbyteoff = (unsigned)cgc * 4u;
    const bool active = (tid >= 7) && (tid < 7 + OUTW) && (cg < WIDTH);

    // 1/(7*k^2) for k = 3,5,7,9,11,13,15
    const float coef[7] = { 1.0f/63.0f, 1.0f/175.0f, 1.0f/343.0f, 1.0f/567.0f,
                            1.0f/847.0f, 1.0f/1183.0f, 1.0f/1575.0f };

    // ---- prefill rolling buffer with rows r0-8 .. r0+6 (row mod 16 == slot,
    //      static because r0 % 16 == 0 and loops are fully unrolled) ----
    float buf[16];
    #pragma unroll
    for (int j = 0; j < 15; ++j) {
        int rr = r0 - 8 + j;                       // rr mod 16 == (j+8)&15
        buf[(j + 8) & 15] = x[(size_t)clampi(rr, 0, HEIGHT - 1) * WIDTH + cgc];
    }

    // ---- init 7 vertical running sums centered on virtual row r0-1 ----
    float rsum[7];
    {
        float s = buf[14] + buf[15] + buf[0];      // rows r0-2, r0-1, r0
        rsum[0] = s;
        #pragma unroll
        for (int p = 2; p <= 7; ++p) {
            s += buf[(15 - p) & 15] + buf[(15 + p) & 15];
            rsum[p - 1] = s;
        }
    }

    // Pre-issue async mask stages for rows r0, r0+1 (step parity 0).
    async_f32_to_lds(mask + (size_t)r0 * WIDTH,       mbyteoff, &Mlds[0][0][tid]);
    async_f32_to_lds(mask + (size_t)(r0 + 1) * WIDTH, mbyteoff, &Mlds[0][1][tid]);

    for (int rb = 0; rb < RH; rb += 16) {
        #pragma unroll
        for (int u = 0; u < 16; u += 2) {
            const int r  = r0 + rb + u;            // rows r, r+1 ; r mod 16 == u
            const int sp = (u >> 1) & 1;           // step parity
            const int so = sp ^ 1;

            // Stage masks for the NEXT step (rows r+2, r+3) -> other parity.
            int rn2 = r + 2; if (rn2 > rmax) rn2 = rmax;
            int rn3 = r + 3; if (rn3 > rmax) rn3 = rmax;
            async_f32_to_lds(mask + (size_t)rn2 * WIDTH, mbyteoff, &Mlds[so][0][tid]);
            async_f32_to_lds(mask + (size_t)rn3 * WIDTH, mbyteoff, &Mlds[so][1][tid]);

            // Issue both leading x loads early; prefetch 16 rows ahead.
            const float nv0 = x[(size_t)clampi(r + 7, 0, HEIGHT - 1) * WIDTH + cgc];
            const float nv1 = x[(size_t)clampi(r + 8, 0, HEIGHT - 1) * WIDTH + cgc];
            __builtin_prefetch(&x[(size_t)clampi(r + 23, 0, HEIGHT - 1) * WIDTH + cgc], 0, 1);
            __builtin_prefetch(&x[(size_t)clampi(r + 24, 0, HEIGHT - 1) * WIDTH + cgc], 0, 1);

            // ---- row r ----  (consumes old slot (u+8)&15 == row r-8 BEFORE
            //                   it is overwritten with row r+8)
            buf[(u + 7) & 15] = nv0;
            #pragma unroll
            for (int p = 1; p <= 7; ++p)
                rsum[p - 1] += buf[(u + p) & 15] - buf[(u - 1 - p + 32) & 15];
            {
                float t = 0.0f;
                #pragma unroll
                for (int m = 7; m >= 1; --m) {
                    t += rsum[m - 1] * coef[m - 1];
                    Tlds[sp][0][m - 1][tid + 8] = t;
                }
            }

            // ---- row r+1 ----
            buf[(u + 8) & 15] = nv1;
            #pragma unroll
            for (int p = 1; p <= 7; ++p)
                rsum[p - 1] += buf[(u + 1 + p) & 15] - buf[(u - p + 32) & 15];
            {
                float t = 0.0f;
                #pragma unroll
                for (int m = 7; m >= 1; --m) {
                    t += rsum[m - 1] * coef[m - 1];
                    Tlds[sp][1][m - 1][tid + 8] = t;
                }
            }

            __syncthreads();

            // Retire this step's two mask stages (2 newer ones stay in flight).
            WAIT_ASYNCCNT(2);
            const float mval0 = Mlds[sp][0][tid];
            const float mval1 = Mlds[sp][1][tid];

            // out(c) = T1(c) + sum_{j=1..7} [ T_j(c-j) + T_j(c+j) ]
            float acc0 = Tlds[sp][0][0][tid + 8];
            float acc1 = Tlds[sp][1][0][tid + 8];
            #pragma unroll
            for (int j = 1; j <= 7; ++j) {
                acc0 += Tlds[sp][0][j - 1][tid + 8 - j]
                      + Tlds[sp][0][j - 1][tid + 8 + j];
                acc1 += Tlds[sp][1][j - 1][tid + 8 - j]
                      + Tlds[sp][1][j - 1][tid + 8 + j];
            }

            if (active) {
                out[(size_t)r       * WIDTH + cg] = acc0 * mval0;
                out[(size_t)(r + 1) * WIDTH + cg] = acc1 * mval1;
            }
        }
    }
}

extern "C" void kernel_launch(void* const* d_in, const int* in_sizes, int n_in,
                              void* d_out, int out_size, void* d_ws, size_t ws_size,
                              hipStream_t stream) {
    (void)in_sizes; (void)n_in; (void)out_size; (void)d_ws; (void)ws_size;
    const float* x    = (const float*)d_in[0];
    const float* mask = (const float*)d_in[1];
    float*       out  = (float*)d_out;

    dim3 grid((WIDTH + OUTW - 1) / OUTW, HEIGHT / RH);   // 17 x 32 = 544 blocks
    dim3 block(NTHR);
    msbox7_kernel<<<grid, block, 0, stream>>>(x, mask, out);
}